// CombinedLoss_67602785239088
// MI455X (gfx1250) — compile-verified
//
#include <hip/hip_runtime.h>

typedef __attribute__((ext_vector_type(2))) float v2f;
typedef __attribute__((ext_vector_type(8))) float v8f;

// One wave = one 16-row strip of X for one batch. Loops over all Y points in
// 16-column tiles; each tile is one V_WMMA_F32_16X16X4_F32:
//   A row m = [x0, x1, x2, 1],  B col p = [-2y0, -2y1, -2y2, |y_p|^2],  C = 0
//   => D[m,p] = |y_p|^2 - 2 x_m.y_p        (row-constant |x_m|^2 added after
//      the min, since it commutes with min over p).
__global__ __launch_bounds__(256) void chamfer_rowmin_kernel(
    const float* __restrict__ X, const float* __restrict__ Y,
    float* __restrict__ partials, int N, int stripsPerBatch)
{
    const int lane = threadIdx.x & 31;
    const int wave = blockIdx.x * (blockDim.x >> 5) + (threadIdx.x >> 5);
    const int b     = wave / stripsPerBatch;
    const int strip = wave - b * stripsPerBatch;
    const int half  = lane >> 4;   // 0: lanes 0-15 (K=0,1), 1: lanes 16-31 (K=2,3)
    const int l16   = lane & 15;

    // Load this lane's x point (row m = strip*16 + l16; same point both halves).
    const int m = strip * 16 + l16;
    const float* xp = X + ((size_t)b * N + (size_t)m) * 3;
    const float x0 = xp[0], x1 = xp[1], x2 = xp[2];
    const float x2sum = x0 * x0 + x1 * x1 + x2 * x2;

    // 32-bit A-matrix 16x4 layout: lanes 0-15 -> VGPR0=K0, VGPR1=K1;
    // lanes 16-31 -> VGPR0=K2, VGPR1=K3.  A = [x0, x1, x2, 1].
    v2f a;
    a.x = half ? x2 : x0;
    a.y = half ? 1.0f : x1;

    // C-matrix is constant zero -> loop-invariant (inline 0 / hoisted).
    v8f czero;
#pragma unroll
    for (int i = 0; i < 8; ++i) czero[i] = 0.0f;

    v8f rowmin;
#pragma unroll
    for (int i = 0; i < 8; ++i) rowmin[i] = 3.0e38f;

    const float* Yb = Y + (size_t)b * (size_t)N * 3;
    const int ptiles = N >> 4;
#pragma unroll 4
    for (int pt = 0; pt < ptiles; ++pt) {
        const int n = (pt << 4) + l16;           // column p handled by this lane
        const float* yp = Yb + (size_t)n * 3;
        const float y0 = yp[0], y1 = yp[1], y2 = yp[2];
        const float y2sum = y0 * y0 + y1 * y1 + y2 * y2;

        // B-matrix 4x16 mirrors the A layout: B = [-2y0, -2y1, -2y2, |y|^2].
        v2f bm;
        bm.x = half ? (-2.0f * y2) : (-2.0f * y0);
        bm.y = half ? y2sum : (-2.0f * y1);

        // D = A x B + 0 : 16x16 tile of (|y_p|^2 - 2 x_m.y_p).
        v8f d = __builtin_amdgcn_wmma_f32_16x16x4_f32(
            false, a, false, bm, (short)0, czero, false, false);

#pragma unroll
        for (int i = 0; i < 8; ++i) rowmin[i] = fminf(rowmin[i], d[i]);
    }

    // Min across the 16 columns held by each 16-lane half.
    // D layout: VGPR i, lanes 0-15 -> row strip*16+i; lanes 16-31 -> +8+i.
#pragma unroll
    for (int off = 8; off >= 1; off >>= 1) {
#pragma unroll
        for (int i = 0; i < 8; ++i)
            rowmin[i] = fminf(rowmin[i], __shfl_xor(rowmin[i], off, 32));
    }

    // Sum the 16 per-row minima of this strip (components 0..7 = rows 0..7 in
    // the lower half, rows 8..15 in the upper half).
    float s = 0.0f;
#pragma unroll
    for (int i = 0; i < 8; ++i) s += rowmin[i];
    s += __shfl_xor(s, 16, 32);                  // combine halves

    // Add back the row-constant |x_m|^2 terms: each 16-lane half holds all 16
    // rows' x2sum, so an intra-half tree sum yields sum_m |x_m|^2.
    float sx = x2sum;
#pragma unroll
    for (int off = 8; off >= 1; off >>= 1)
        sx += __shfl_xor(sx, off, 32);
    s += sx;

    if (lane == 0) partials[wave] = s;
}

// Deterministic fixed-order final reduction: out = scale * sum(partials).
__global__ __launch_bounds__(256) void chamfer_reduce_kernel(
    const float* __restrict__ partials, float* __restrict__ out,
    int count, float scale)
{
    __shared__ float sm[256];
    float s = 0.0f;
    for (int i = threadIdx.x; i < count; i += 256) s += partials[i];
    sm[threadIdx.x] = s;
    __syncthreads();
    for (int off = 128; off >= 1; off >>= 1) {
        if ((int)threadIdx.x < off) sm[threadIdx.x] += sm[threadIdx.x + off];
        __syncthreads();
    }
    if (threadIdx.x == 0) out[0] = sm[0] * scale;
}

extern "C" void kernel_launch(void* const* d_in, const int* in_sizes, int n_in,
                              void* d_out, int out_size, void* d_ws, size_t ws_size,
                              hipStream_t stream) {
    const float* gts  = (const float*)d_in[0];   // gts_X  [B,N,3]
    const float* pred = (const float*)d_in[1];   // pred_X [B,N,3]
    // d_in[2] (gts_normals) unused by the reference path.
    float* ws  = (float*)d_ws;
    float* out = (float*)d_out;

    const int B = 4, D = 3;
    const int N = in_sizes[0] / (B * D);         // 8192
    const int strips      = N / 16;              // 512
    const int wavesPerDir = B * strips;          // 2048 waves per direction
    const int blocks      = (wavesPerDir * 32) / 256;  // 8 waves / block

    // cham_x: min over pred for each gts point.
    chamfer_rowmin_kernel<<<blocks, 256, 0, stream>>>(gts, pred, ws, N, strips);
    // cham_y: same kernel, roles swapped (column-min == row-min of transpose).
    chamfer_rowmin_kernel<<<blocks, 256, 0, stream>>>(pred, gts, ws + wavesPerDir, N, strips);

    // mean_b(mean_n rowmin_x) + mean_b(mean_p rowmin_y) with N==P:
    // out = (sum of all partials) / (B*N)
    chamfer_reduce_kernel<<<1, 256, 0, stream>>>(ws, out, 2 * wavesPerDir,
                                                 1.0f / (float)(B * N));
}